// NodeModel_21552145891503
// MI455X (gfx1250) — compile-verified
//
#include <hip/hip_runtime.h>
#include <hip/hip_bf16.h>

typedef __attribute__((ext_vector_type(2))) float v2f;
typedef __attribute__((ext_vector_type(8))) float v8f;

// ---------------------------------------------------------------------------
// Kernel 1: zero the agg accumulator (d_ws), needed fresh on every call.
// ---------------------------------------------------------------------------
__global__ void zero_f32_kernel(float* __restrict__ p, long long n) {
    long long i = (long long)blockIdx.x * blockDim.x + threadIdx.x;
    if (i < n) p[i] = 0.0f;
}

// ---------------------------------------------------------------------------
// Kernel 2: edge scatter: agg[col] += x[row], one thread per edge.
// edge_index is int64 [2, E] row-major: rows at ei[0..E), cols at ei[E..2E).
// Streaming index reads dominate HBM traffic; x/agg live in 192MB L2.
// ---------------------------------------------------------------------------
__global__ void edge_scatter_kernel(const long long* __restrict__ ei,
                                    const float* __restrict__ x,
                                    float* __restrict__ agg,
                                    long long n_edges) {
    long long e = (long long)blockIdx.x * blockDim.x + threadIdx.x;
    if (e >= n_edges) return;

    // prefetch the streaming index arrays 8192 entries (64 KB) ahead (one lane per line)
    const long long PF = 8192;
    if (((threadIdx.x & 15) == 0) && (e + PF < n_edges)) {
        __builtin_prefetch(&ei[e + PF], 0, 1);
        __builtin_prefetch(&ei[n_edges + e + PF], 0, 1);
    }

    long long row = ei[e];
    long long col = ei[n_edges + e];
    const float* xs = x + row * 3;
    float x0 = xs[0], x1 = xs[1], x2 = xs[2];
    float* ap = agg + col * 3;
    atomicAdd(ap + 0, x0);
    atomicAdd(ap + 1, x1);
    atomicAdd(ap + 2, x2);
}

// ---------------------------------------------------------------------------
// Kernel 3: per-node MLP via V_WMMA_F32_16X16X4_F32. 16 nodes per wave32.
//   H0 = [x | agg]            (16x6)
//   H1 = relu(H0 @ W1 + b1)   (16x16)
//   H2 = relu(H1 @ W2 + b2)   (16x16)
//   OUT = H2 @ W3 + b3        (16x3)
// D-layout -> A-layout relayout between layers goes through a 16x17 LDS tile.
// ---------------------------------------------------------------------------
__device__ __forceinline__ v8f wmma_f32_k4(v2f a, v2f b, v8f c) {
    return __builtin_amdgcn_wmma_f32_16x16x4_f32(
        /*neg_a=*/false, a, /*neg_b=*/false, b,
        /*c_mod=*/(short)0, c, /*reuse_a=*/false, /*reuse_b=*/false);
}

__device__ __forceinline__ void lds_wave_fence() {
    __builtin_amdgcn_wave_barrier();
    asm volatile("s_wait_dscnt 0" ::: "memory");
    __builtin_amdgcn_wave_barrier();
}

#define WAVES_PER_BLOCK 8

__global__ void mlp_wmma_kernel(const float* __restrict__ x,
                                const float* __restrict__ agg,
                                const float* __restrict__ W1, const float* __restrict__ b1,
                                const float* __restrict__ W2, const float* __restrict__ b2,
                                const float* __restrict__ W3, const float* __restrict__ b3,
                                float* __restrict__ out,
                                long long n_nodes) {
    __shared__ float smem[WAVES_PER_BLOCK][16 * 17];

    const int wave = threadIdx.x >> 5;
    const int lane = threadIdx.x & 31;
    const int m    = lane & 15;   // A row / B col / D col index
    const int half = lane >> 4;   // lane half selects K/M sub-range

    const long long n_tiles = (n_nodes + 15) >> 4;
    const long long tile = (long long)blockIdx.x * WAVES_PER_BLOCK + wave;
    if (tile >= n_tiles) return;  // uniform per wave -> EXEC all-1s below
    const long long base = tile << 4;

    float* sm = smem[wave];

    // ---- load weight B-fragments (B frag j: rows 4j+2*half+{0,1}, col m) ----
    v2f bw1[2];
#pragma unroll
    for (int j = 0; j < 2; ++j) {
        int r0 = 4 * j + 2 * half;
        bw1[j][0] = (r0     < 6) ? W1[(r0)     * 16 + m] : 0.0f;
        bw1[j][1] = (r0 + 1 < 6) ? W1[(r0 + 1) * 16 + m] : 0.0f;
    }
    v2f bw2[4];
#pragma unroll
    for (int j = 0; j < 4; ++j) {
        int r0 = 4 * j + 2 * half;
        bw2[j][0] = W2[(r0)     * 16 + m];
        bw2[j][1] = W2[(r0 + 1) * 16 + m];
    }
    v2f bw3[4];   // W3 is [16][3]; valid cols n<3
#pragma unroll
    for (int j = 0; j < 4; ++j) {
        int r0 = 4 * j + 2 * half;
        bw3[j][0] = (m < 3) ? W3[(r0)     * 3 + m] : 0.0f;
        bw3[j][1] = (m < 3) ? W3[(r0 + 1) * 3 + m] : 0.0f;
    }
    const float bias1 = b1[m];
    const float bias2 = b2[m];
    const float bias3 = (m < 3) ? b3[m] : 0.0f;

    // ---- layer 1 A-fragments directly from global: H0[m][k] ----
    long long node = base + m;
    if (node >= n_nodes) node = n_nodes - 1;   // data clamp only; EXEC unchanged
    const float* xp = x + node * 3;
    const float* ap = agg + node * 3;
    v2f a0, a1;
    if (half == 0) { a0[0] = xp[0]; a0[1] = xp[1]; a1[0] = ap[1]; a1[1] = ap[2]; }
    else           { a0[0] = xp[2]; a0[1] = ap[0]; a1[0] = 0.0f;  a1[1] = 0.0f;  }

    v8f acc;
#pragma unroll
    for (int v = 0; v < 8; ++v) acc[v] = bias1;
    acc = wmma_f32_k4(a0, bw1[0], acc);
    acc = wmma_f32_k4(a1, bw1[1], acc);
#pragma unroll
    for (int v = 0; v < 8; ++v) acc[v] = fmaxf(acc[v], 0.0f);

    // ---- relayout D->A via LDS, layer 2 ----
#pragma unroll
    for (int v = 0; v < 8; ++v) sm[(v + 8 * half) * 17 + m] = acc[v];
    lds_wave_fence();
    v2f a2[4];
#pragma unroll
    for (int j = 0; j < 4; ++j) {
        a2[j][0] = sm[m * 17 + 4 * j + 2 * half];
        a2[j][1] = sm[m * 17 + 4 * j + 2 * half + 1];
    }
    lds_wave_fence();

#pragma unroll
    for (int v = 0; v < 8; ++v) acc[v] = bias2;
#pragma unroll
    for (int j = 0; j < 4; ++j) acc = wmma_f32_k4(a2[j], bw2[j], acc);
#pragma unroll
    for (int v = 0; v < 8; ++v) acc[v] = fmaxf(acc[v], 0.0f);

    // ---- relayout, layer 3 ----
#pragma unroll
    for (int v = 0; v < 8; ++v) sm[(v + 8 * half) * 17 + m] = acc[v];
    lds_wave_fence();
    v2f a3[4];
#pragma unroll
    for (int j = 0; j < 4; ++j) {
        a3[j][0] = sm[m * 17 + 4 * j + 2 * half];
        a3[j][1] = sm[m * 17 + 4 * j + 2 * half + 1];
    }
    lds_wave_fence();

#pragma unroll
    for (int v = 0; v < 8; ++v) acc[v] = bias3;
#pragma unroll
    for (int j = 0; j < 4; ++j) acc = wmma_f32_k4(a3[j], bw3[j], acc);

    // ---- write OUT[16 nodes][3]: stage through LDS, then linear stores ----
#pragma unroll
    for (int v = 0; v < 8; ++v) sm[(v + 8 * half) * 17 + m] = acc[v];
    lds_wave_fence();
    for (int i = lane; i < 48; i += 32) {
        int r = i / 3, c = i - 3 * r;
        long long nd = base + r;
        if (nd < n_nodes) out[nd * 3 + c] = sm[r * 17 + c];
    }
}

// ---------------------------------------------------------------------------
// Host launcher
// ---------------------------------------------------------------------------
extern "C" void kernel_launch(void* const* d_in, const int* in_sizes, int n_in,
                              void* d_out, int out_size, void* d_ws, size_t ws_size,
                              hipStream_t stream) {
    const float*     x  = (const float*)d_in[0];
    const long long* ei = (const long long*)d_in[1];
    // d_in[2] edge_attr, d_in[3] u, d_in[4] batch: unused by the reference math
    const float* W1 = (const float*)d_in[5];
    const float* b1 = (const float*)d_in[6];
    const float* W2 = (const float*)d_in[7];
    const float* b2 = (const float*)d_in[8];
    const float* W3 = (const float*)d_in[9];
    const float* b3 = (const float*)d_in[10];
    float* out = (float*)d_out;
    float* agg = (float*)d_ws;   // [n_nodes * 3] fp32 accumulator

    const long long n_nodes = (long long)in_sizes[0] / 3;
    const long long n_edges = (long long)in_sizes[1] / 2;

    // 1) zero agg
    {
        long long n = n_nodes * 3;
        int threads = 256;
        long long blocks = (n + threads - 1) / threads;
        zero_f32_kernel<<<(unsigned)blocks, threads, 0, stream>>>(agg, n);
    }
    // 2) edge scatter
    {
        int threads = 256;
        long long blocks = (n_edges + threads - 1) / threads;
        edge_scatter_kernel<<<(unsigned)blocks, threads, 0, stream>>>(ei, x, agg, n_edges);
    }
    // 3) WMMA MLP
    {
        long long n_tiles = (n_nodes + 15) / 16;
        long long blocks = (n_tiles + WAVES_PER_BLOCK - 1) / WAVES_PER_BLOCK;
        mlp_wmma_kernel<<<(unsigned)blocks, 32 * WAVES_PER_BLOCK, 0, stream>>>(
            x, agg, W1, b1, W2, b2, W3, b3, out, n_nodes);
    }
}